// ReadConvolverHybridDNN_18219251269831
// MI455X (gfx1250) — compile-verified
//
#include <hip/hip_runtime.h>

typedef __attribute__((ext_vector_type(2))) float v2f;
typedef __attribute__((ext_vector_type(8))) float v8f;

#define C_IN  12
#define L_WIN 32
#define D_HID 32
#define NOUT_ 2

// ---------------------------------------------------------------------------
// Exclusive prefix scan of the two ragged read-count arrays (A elements each)
// into workspace. Single block, Hillis-Steele in LDS with running carry.
// ---------------------------------------------------------------------------
__global__ __launch_bounds__(1024) void
exscan_kernel(const int* __restrict__ c0, const int* __restrict__ c1,
              int* __restrict__ o0, int* __restrict__ o1, int nA)
{
    __shared__ int buf[1024];
    __shared__ int carry_s;
    const int t = threadIdx.x;

    for (int src = 0; src < 2; ++src) {
        const int* __restrict__ c = src ? c1 : c0;
        int* __restrict__ o       = src ? o1 : o0;
        if (t == 0) carry_s = 0;
        __syncthreads();
        for (int base = 0; base < nA; base += 1024) {
            const int v = (base + t < nA) ? c[base + t] : 0;
            buf[t] = v;
            __syncthreads();
            #pragma unroll
            for (int d = 1; d < 1024; d <<= 1) {
                const int add = (t >= d) ? buf[t - d] : 0;
                __syncthreads();
                buf[t] += add;
                __syncthreads();
            }
            const int incl  = buf[t];
            const int carry = carry_s;
            if (base + t < nA) o[base + t] = carry + incl - v;   // exclusive
            __syncthreads();
            if (t == 1023) carry_s = carry + incl;
            __syncthreads();
        }
        __syncthreads();
    }
}

// ---------------------------------------------------------------------------
// One block per allele; wave 0 handles source 0, wave 1 handles source 1.
// Per read r: F^T[l][d] = relu( (T_r^T W^T)[l][d] + b[d] ) via chained
// V_WMMA_F32_16X16X4_F32 (M=l tile, N=d tile, K=c in 3 steps of 4).
// C/D layout puts l in the VGPR rows, so sum-over-l is in-lane: ReLU is a
// single v_med3_f32 (med3(x, 0, +inf) == max(x, 0), no canonicalize op) and
// accumulation goes into per-slot v8f accumulators so pk_add pairs need no
// register shuffling. The horizontal 8->1 sum + cross-half shfl happen once
// per allele, after the read loop. Then normalize by depth*L and apply the
// 64->2 head through LDS.
// ---------------------------------------------------------------------------
__global__ __launch_bounds__(64) void
readconv_site_kernel(const float* __restrict__ t0, const float* __restrict__ t1,
                     const float* __restrict__ W0, const float* __restrict__ b0,
                     const float* __restrict__ W1, const float* __restrict__ b1,
                     const float* __restrict__ W2, const float* __restrict__ b2,
                     const int* __restrict__ cnt0, const int* __restrict__ cnt1,
                     const float* __restrict__ dm0, const float* __restrict__ dm1,
                     const int* __restrict__ off0, const int* __restrict__ off1,
                     float* __restrict__ out)
{
    __shared__ float sm[2 * D_HID];

    const int a    = blockIdx.x;
    const int w    = threadIdx.x >> 5;   // wave id == data source (uniform/wave)
    const int lane = threadIdx.x & 31;
    const int lo   = lane & 15;
    const int hi   = lane >> 4;

    const float* __restrict__ T  = w ? t1   : t0;
    const float* __restrict__ Wm = w ? W1   : W0;
    const float* __restrict__ Bv = w ? b1   : b0;
    const int*   __restrict__ cn = w ? cnt1 : cnt0;
    const int*   __restrict__ of = w ? off1 : off0;
    const float* __restrict__ dm = w ? dm1  : dm0;

    // B operand = W^T, constant across reads.
    // B[k][n]: lane -> n = nt*16+lo ; VGPR j -> k = ks*4 + 2*hi + j.
    v2f wb[2][3];
    #pragma unroll
    for (int nt = 0; nt < 2; ++nt) {
        #pragma unroll
        for (int ks = 0; ks < 3; ++ks) {
            const int d = nt * 16 + lo;
            const int c = ks * 4 + 2 * hi;
            wb[nt][ks].x = Wm[d * C_IN + c];
            wb[nt][ks].y = Wm[d * C_IN + c + 1];
        }
    }
    const float bias0 = Bv[lo];        // d = lo        (nt = 0)
    const float bias1 = Bv[16 + lo];   // d = 16 + lo   (nt = 1)

    const int n   = cn[a];
    const int off = of[a];

    const float kInf = __builtin_inff();

    // Per-slot accumulators of sum_{reads,l} relu(...) for this lane's d.
    v8f vacc0 = {0.f, 0.f, 0.f, 0.f, 0.f, 0.f, 0.f, 0.f};
    v8f vacc1 = vacc0;

    #pragma unroll 2
    for (int i = 0; i < n; ++i) {
        const float* __restrict__ tr = T + (size_t)(off + i) * (C_IN * L_WIN);

        // Accumulators seeded with bias (every l row gets +b[d]).
        v8f c00 = {bias0, bias0, bias0, bias0, bias0, bias0, bias0, bias0};
        v8f c01 = {bias1, bias1, bias1, bias1, bias1, bias1, bias1, bias1};
        v8f c10 = c00;
        v8f c11 = c01;

        #pragma unroll
        for (int ks = 0; ks < 3; ++ks) {
            const int c = ks * 4 + 2 * hi;
            // A operand = T_r^T: lane -> m = l (mod 16); VGPR j -> k = c + j.
            v2f a0, a1;
            a0.x = tr[c * L_WIN + lo];            // l = lo       (lt = 0)
            a0.y = tr[(c + 1) * L_WIN + lo];
            a1.x = tr[c * L_WIN + 16 + lo];       // l = 16 + lo  (lt = 1)
            a1.y = tr[(c + 1) * L_WIN + 16 + lo];

            c00 = __builtin_amdgcn_wmma_f32_16x16x4_f32(false, a0, false, wb[0][ks],
                                                        (short)0, c00, false, false);
            c01 = __builtin_amdgcn_wmma_f32_16x16x4_f32(false, a0, false, wb[1][ks],
                                                        (short)0, c01, false, false);
            c10 = __builtin_amdgcn_wmma_f32_16x16x4_f32(false, a1, false, wb[0][ks],
                                                        (short)0, c10, false, false);
            c11 = __builtin_amdgcn_wmma_f32_16x16x4_f32(false, a1, false, wb[1][ks],
                                                        (short)0, c11, false, false);
        }

        // relu(x) = med3(x, 0, +inf): one VALU op per element; accumulate in
        // per-slot vector accumulators (adjacent pairs -> clean v_pk_add).
        #pragma unroll
        for (int v = 0; v < 8; ++v) {
            vacc0[v] += __builtin_amdgcn_fmed3f(c00[v], 0.f, kInf)
                      + __builtin_amdgcn_fmed3f(c10[v], 0.f, kInf);
            vacc1[v] += __builtin_amdgcn_fmed3f(c01[v], 0.f, kInf)
                      + __builtin_amdgcn_fmed3f(c11[v], 0.f, kInf);
        }
    }

    // Horizontal 8->1 sum of the per-slot accumulators (once per allele).
    float acc0 = ((vacc0[0] + vacc0[1]) + (vacc0[2] + vacc0[3]))
               + ((vacc0[4] + vacc0[5]) + (vacc0[6] + vacc0[7]));
    float acc1 = ((vacc1[0] + vacc1[1]) + (vacc1[2] + vacc1[3]))
               + ((vacc1[4] + vacc1[5]) + (vacc1[6] + vacc1[7]));

    // Complete the l-sum across the two lane halves (one shuffle per d-tile).
    acc0 += __shfl_xor(acc0, 16, 32);
    acc1 += __shfl_xor(acc1, 16, 32);

    // segment-sum / depth, then mean over L.
    const float scale = 1.f / (dm[a] * (float)L_WIN);
    if (hi == 0) {
        sm[w * D_HID + lo]      = acc0 * scale;   // concat dims [w*32 .. w*32+31]
        sm[w * D_HID + 16 + lo] = acc1 * scale;
    }
    __syncthreads();

    // Head: out[a][j] = b2[j] + sum_{k<64} W2[j][k] * pooled[k]   (wave 0 only)
    if (threadIdx.x < 32) {
        const float v0 = sm[lane];
        const float v1 = sm[32 + lane];
        #pragma unroll
        for (int j = 0; j < NOUT_; ++j) {
            float p = W2[j * 64 + lane] * v0 + W2[j * 64 + 32 + lane] * v1;
            #pragma unroll
            for (int m = 16; m >= 1; m >>= 1) p += __shfl_xor(p, m, 32);
            if (lane == 0) out[a * NOUT_ + j] = p + b2[j];
        }
    }
}

// ---------------------------------------------------------------------------
extern "C" void kernel_launch(void* const* d_in, const int* in_sizes, int n_in,
                              void* d_out, int out_size, void* d_ws, size_t ws_size,
                              hipStream_t stream)
{
    const float* t0  = (const float*)d_in[0];
    const float* t1  = (const float*)d_in[1];
    const float* W0  = (const float*)d_in[2];
    const float* b0  = (const float*)d_in[3];
    const float* W1  = (const float*)d_in[4];
    const float* b1  = (const float*)d_in[5];
    const float* W2  = (const float*)d_in[6];
    const float* b2  = (const float*)d_in[7];
    const int*   cn0 = (const int*)d_in[8];
    const int*   cn1 = (const int*)d_in[9];
    // d_in[10] = numAllelesPerSite (unused by the reference math)
    const float* dm0 = (const float*)d_in[11];
    const float* dm1 = (const float*)d_in[12];

    const int nA = in_sizes[8];

    int* off0 = (int*)d_ws;
    int* off1 = off0 + nA;

    exscan_kernel<<<1, 1024, 0, stream>>>(cn0, cn1, off0, off1, nA);

    readconv_site_kernel<<<nA, 64, 0, stream>>>(
        t0, t1, W0, b0, W1, b1, W2, b2,
        cn0, cn1, dm0, dm1, off0, off1, (float*)d_out);
}